// SimpleGCN_66838281061190
// MI455X (gfx1250) — compile-verified
//
#include <hip/hip_runtime.h>
#include <cstdint>

#define N_NODES 50000
#define N_EDGES 800000
#define IN_DIM 512
#define HID 128
#define MTILES (N_NODES / 16)   // 3125, exact

#define AROW_STRIDE 133         // 128 data + 5 pad DW  (TDM pad: interval=128DW, amount=5DW)
#define WROW_STRIDE 144         // 128 data + 16 pad DW (TDM pad: interval=128DW, amount=16DW)

typedef __attribute__((ext_vector_type(2))) float v2f;
typedef __attribute__((ext_vector_type(8))) float v8f;
typedef __attribute__((ext_vector_type(4))) unsigned int u32x4;
typedef __attribute__((ext_vector_type(8))) int i32x8;
typedef __attribute__((ext_vector_type(4))) int i32x4;

// ---- Tensor Data Mover: 2-D f32 tile Global -> LDS with optional row padding ----
// D# per CDNA5 ISA ch.8: group0 {count,lds_addr,global_addr,type=2},
// group1 {data_size=4B, pad ctl, tensor_dim0/1, tile_dim0/1, dim0_stride},
// groups 2/3 (and extra group) zero => 2-D tile (tile_dim2/3 = 0 -> unused).
__device__ __forceinline__ void tdm_load_2d(const float* gsrc, const float* lds_dst,
                                            int width, int tile_rows, int valid_rows,
                                            int gstride, int pad_interval_code,
                                            int pad_amount_code) {
  unsigned long long ga = (unsigned long long)(uintptr_t)gsrc;
  unsigned lds = (unsigned)(uintptr_t)lds_dst;          // addr[31:0] = LDS byte offset
  u32x4 g0;
  g0[0] = 1u;                                           // count=1, user mode, no gather
  g0[1] = lds;                                          // lds_addr
  g0[2] = (unsigned)ga;                                 // global_addr[31:0]
  g0[3] = (unsigned)((ga >> 32) & 0x01ffffffull) | 0x80000000u;  // ga[56:32] | type=2
  i32x8 g1;
  g1[0] = (int)((2u << 16) | (1u << 20) |               // data_size=4B, pad_enable
                ((unsigned)pad_interval_code << 22) |
                ((unsigned)pad_amount_code << 25));
  g1[1] = (int)(((unsigned)width & 0xffffu) << 16);                     // tensor_dim0 lo
  g1[2] = (int)((((unsigned)width >> 16) & 0xffffu) |
                (((unsigned)valid_rows & 0xffffu) << 16));              // dim0 hi | dim1 lo
  g1[3] = (int)(((((unsigned)valid_rows >> 16) & 0xffffu)) |
                (((unsigned)width & 0xffffu) << 16));                   // dim1 hi | tile_dim0
  g1[4] = (int)((unsigned)tile_rows & 0xffffu);                         // tile_dim1, tile_dim2=0
  g1[5] = gstride;                                                      // dim0_stride lo32
  g1[6] = 0;
  g1[7] = 0;
  i32x4 z4 = (i32x4)0;
  i32x8 z8 = (i32x8)0;
  __builtin_amdgcn_tensor_load_to_lds(g0, g1, z4, z4, z8, 0);
}

// ---------------- degree / normalization ----------------
__global__ void k_deg_init(float* __restrict__ deg) {
  int n = blockIdx.x * 256 + threadIdx.x;
  if (n < N_NODES) deg[n] = 1.0f;            // self-loop contributes 1
}

__global__ void k_deg_accum(const int* __restrict__ dst, float* __restrict__ deg) {
  int e = blockIdx.x * 256 + threadIdx.x;
  if (e < N_EDGES) atomicAdd(deg + dst[e], 1.0f);
}

__global__ void k_dinv(const float* __restrict__ deg, float* __restrict__ dinv) {
  int n = blockIdx.x * 256 + threadIdx.x;
  if (n < N_NODES) dinv[n] = rsqrtf(deg[n]); // deg >= 1 always
}

// ------- GEMM1: h = x @ W_gcn  (TDM-staged LDS tiles + f32 WMMA 16x16x4) -------
__global__ __launch_bounds__(256) void k_gemm1(const float* __restrict__ x,
                                               const float* __restrict__ Wg,
                                               float* __restrict__ h) {
  __shared__ float ldsA[128 * AROW_STRIDE];  // 128 rows of x, K-slab 128 wide
  __shared__ float ldsW[128 * WROW_STRIDE];  // 128 K-rows of W, N=128 wide
  const int lane = threadIdx.x & 31;
  const int wave = threadIdx.x >> 5;
  const int tileM = blockIdx.x * 8 + wave;
  const int ml = lane & 15;                  // A row / B,D col within tile
  const int kh = (lane >> 4) << 1;           // K sub-offset: 0 or 2
  const int rowsLeft = N_NODES - blockIdx.x * 128;
  const int rowsValid = rowsLeft < 128 ? rowsLeft : 128;
  v8f acc[8] = {};
  for (int ks = 0; ks < IN_DIM; ks += 128) {
    __syncthreads();                         // previous slab fully consumed
    if (wave == 0) {                         // one TDM issue per block (EXEC ignored)
      tdm_load_2d(x + (size_t)blockIdx.x * 128 * IN_DIM + ks, ldsA,
                  128, 128, rowsValid, IN_DIM, /*interval 128DW*/6, /*amount 5DW*/4);
      tdm_load_2d(Wg + (size_t)ks * HID, ldsW,
                  128, 128, 128, HID, /*interval 128DW*/6, /*amount 16DW*/15);
      __builtin_amdgcn_s_wait_tensorcnt(0);
    }
    __syncthreads();                         // publish LDS tiles
    if (tileM < MTILES) {                    // wave-uniform: EXEC all-ones at WMMA
      const float* ar = ldsA + (wave * 16 + ml) * AROW_STRIDE + kh;
      for (int k = 0; k < 128; k += 4) {
        v2f a;
        a.x = ar[k];                         // two b32 reads, bank-conflict-free (stride 133)
        a.y = ar[k + 1];
        const float* bk = ldsW + (size_t)(k + kh) * WROW_STRIDE + ml;
#pragma unroll
        for (int t = 0; t < 8; ++t) {
          v2f b;
          b.x = bk[t * 16];
          b.y = bk[t * 16 + WROW_STRIDE];
          acc[t] = __builtin_amdgcn_wmma_f32_16x16x4_f32(
              false, a, false, b, (short)0, acc[t], false, false);
        }
      }
    }
  }
  if (tileM >= MTILES) return;
  const int n0 = lane & 15;
  const int mb = (lane >> 4) * 8;
  float* hb = h + (size_t)(tileM * 16 + mb) * HID + n0;
#pragma unroll
  for (int v = 0; v < 8; ++v)
#pragma unroll
    for (int t = 0; t < 8; ++t)
      hb[(size_t)v * HID + t * 16] = acc[t][v];
}

// ---------------- aggregation ----------------
__global__ void k_selfloop(const float* __restrict__ h, const float* __restrict__ dinv,
                           float* __restrict__ agg) {
  size_t i = (size_t)blockIdx.x * 256 + threadIdx.x;
  if (i < (size_t)N_NODES * HID) {
    float di = dinv[i >> 7];
    agg[i] = h[i] * di * di;
  }
}

__global__ __launch_bounds__(256) void k_edge_agg(const int* __restrict__ src,
                                                  const int* __restrict__ dst,
                                                  const float* __restrict__ h,
                                                  const float* __restrict__ dinv,
                                                  float* __restrict__ agg) {
  const int e = blockIdx.x * 8 + (threadIdx.x >> 5);   // one wave32 per edge
  const int lane = threadIdx.x & 31;
  if (e >= N_EDGES) return;
  const int s = src[e], d = dst[e];
  const float w = dinv[s] * dinv[d];
  const float4 hv = *(const float4*)(h + (size_t)s * HID + lane * 4);
  float* ap = agg + (size_t)d * HID + lane * 4;        // L2-resident atomics (h/agg << 192MB)
  atomicAdd(ap + 0, hv.x * w);
  atomicAdd(ap + 1, hv.y * w);
  atomicAdd(ap + 2, hv.z * w);
  atomicAdd(ap + 3, hv.w * w);
}

// --- GEMM2: z = relu( relu(agg + b_gcn) @ W1 + b1 ), TDM-staged, fused epilogues ---
__global__ __launch_bounds__(256) void k_gemm2(const float* __restrict__ agg,
                                               const float* __restrict__ bg,
                                               const float* __restrict__ W1,
                                               const float* __restrict__ b1,
                                               float* __restrict__ z) {
  __shared__ float ldsA[128 * AROW_STRIDE];
  __shared__ float ldsW[128 * WROW_STRIDE];
  const int lane = threadIdx.x & 31;
  const int wave = threadIdx.x >> 5;
  const int tileM = blockIdx.x * 8 + wave;
  const int ml = lane & 15;
  const int kh = (lane >> 4) << 1;
  const int rowsLeft = N_NODES - blockIdx.x * 128;
  const int rowsValid = rowsLeft < 128 ? rowsLeft : 128;
  if (wave == 0) {                           // single K-slab: K == 128
    tdm_load_2d(agg + (size_t)blockIdx.x * 128 * HID, ldsA,
                128, 128, rowsValid, HID, 6, 4);
    tdm_load_2d(W1, ldsW, 128, 128, 128, HID, 6, 15);
    __builtin_amdgcn_s_wait_tensorcnt(0);
  }
  __syncthreads();
  v8f acc[8] = {};
  if (tileM < MTILES) {
    const float* ar = ldsA + (wave * 16 + ml) * AROW_STRIDE + kh;
    const float* bgp = bg + kh;
    for (int k = 0; k < HID; k += 4) {
      v2f a;
      a.x = fmaxf(ar[k] + bgp[k], 0.0f);     // fused GCN bias + ReLU on A-load
      a.y = fmaxf(ar[k + 1] + bgp[k + 1], 0.0f);
      const float* bk = ldsW + (size_t)(k + kh) * WROW_STRIDE + ml;
#pragma unroll
      for (int t = 0; t < 8; ++t) {
        v2f b;
        b.x = bk[t * 16];
        b.y = bk[t * 16 + WROW_STRIDE];
        acc[t] = __builtin_amdgcn_wmma_f32_16x16x4_f32(
            false, a, false, b, (short)0, acc[t], false, false);
      }
    }
  }
  if (tileM >= MTILES) return;
  const int n0 = lane & 15;
  const int mb = (lane >> 4) * 8;
  float* zb = z + (size_t)(tileM * 16 + mb) * HID + n0;
#pragma unroll
  for (int v = 0; v < 8; ++v)
#pragma unroll
    for (int t = 0; t < 8; ++t)
      zb[(size_t)v * HID + t * 16] = fmaxf(acc[t][v] + b1[t * 16 + n0], 0.0f);
}

// ---------------- head: y = z @ W2 + b2 ----------------
__global__ void k_head(const float* __restrict__ z, const float* __restrict__ W2,
                       const float* __restrict__ b2, float* __restrict__ y) {
  int n = blockIdx.x * 256 + threadIdx.x;
  if (n >= N_NODES) return;
  const float4* zr = (const float4*)(z + (size_t)n * HID);
  const float4* w  = (const float4*)W2;
  float acc = 0.0f;
#pragma unroll
  for (int i = 0; i < HID / 4; ++i) {
    float4 a = zr[i], b = w[i];
    acc += a.x * b.x + a.y * b.y + a.z * b.z + a.w * b.w;
  }
  y[n] = acc + b2[0];
}

// ---------------- launch ----------------
extern "C" void kernel_launch(void* const* d_in, const int* in_sizes, int n_in,
                              void* d_out, int out_size, void* d_ws, size_t ws_size,
                              hipStream_t stream) {
  const float* x  = (const float*)d_in[0];
  const int*   ei = (const int*)d_in[1];
  const float* Wg = (const float*)d_in[2];
  const float* bg = (const float*)d_in[3];
  const float* W1 = (const float*)d_in[4];
  const float* b1 = (const float*)d_in[5];
  const float* W2 = (const float*)d_in[6];
  const float* b2 = (const float*)d_in[7];
  float* y = (float*)d_out;

  float* h    = (float*)d_ws;                        // N*HID  (25.6 MB)
  float* agg  = h   + (size_t)N_NODES * HID;         // N*HID  (25.6 MB)
  float* dinv = agg + (size_t)N_NODES * HID;         // N
  float* deg  = dinv + N_NODES;                      // N
  const int* src = ei;
  const int* dst = ei + N_EDGES;

  const int B = 256;
  k_deg_init  <<<(N_NODES + B - 1) / B, B, 0, stream>>>(deg);
  k_deg_accum <<<(N_EDGES + B - 1) / B, B, 0, stream>>>(dst, deg);
  k_dinv      <<<(N_NODES + B - 1) / B, B, 0, stream>>>(deg, dinv);

  k_gemm1 <<<(MTILES + 7) / 8, B, 0, stream>>>(x, Wg, h);

  size_t tot = (size_t)N_NODES * HID;
  k_selfloop <<<(unsigned)((tot + B - 1) / B), B, 0, stream>>>(h, dinv, agg);
  k_edge_agg <<<N_EDGES / 8, B, 0, stream>>>(src, dst, h, dinv, agg);

  k_gemm2 <<<(MTILES + 7) / 8, B, 0, stream>>>(agg, bg, W1, b1, h);  // reuse h as z
  k_head  <<<(N_NODES + B - 1) / B, B, 0, stream>>>(h, W2, b2, y);
}